// LaplacianLoss_90211493085474
// MI455X (gfx1250) — compile-verified
//
#include <hip/hip_runtime.h>

// Problem constants (match the reference)
#define NV     10242
#define NBAT   4
#define NT     641                 // ceil(NV/16) n-tiles
#define KSTEPS 2561                // ceil(NV/4) k-steps of 4
#define KFULL  2560                // k-steps where all 4 m are in range
#define KPAD   (KSTEPS * 4)        // 10244, padded K extent of xT

typedef float v2f __attribute__((ext_vector_type(2)));
typedef float v8f __attribute__((ext_vector_type(8)));

#define WMMA_F32X4(A, B, C) \
    __builtin_amdgcn_wmma_f32_16x16x4_f32(false, (A), false, (B), (short)0, (C), false, false)

// ---------------------------------------------------------------------------
// Build zero-padded transposed RHS: xT[j][k] = x[j/3][k][j%3], j in [0,16),
// k in [0,KPAD); rows j>=12 and columns k>=NV are 0 so the GEMM hot loop
// needs no bounds logic on the A operand.
// ---------------------------------------------------------------------------
__global__ void xT_build(const float* __restrict__ x, float* __restrict__ xT) {
    int t = blockIdx.x * blockDim.x + threadIdx.x;
    if (t >= 16 * KPAD) return;
    int j = t / KPAD;
    int k = t - j * KPAD;
    float v = 0.0f;
    if (j < 12 && k < NV) {
        int b = j / 3;
        int d = j - 3 * b;
        v = x[((size_t)b * NV + k) * 3 + d];
    }
    xT[t] = v;
}

// ---------------------------------------------------------------------------
// Main GEMM: one wave per (n-tile, k-chunk). D(16x16) accumulates
// A(16x4 = xT block) x B(4x16 = L^T block) via V_WMMA_F32_16X16X4_F32.
// 8-step batches, double-buffered in registers: 16 b64 loads (4 KB/wave) are
// in flight while the previous batch's 8 WMMAs retire. All control state is
// broadcast to SGPRs via readfirstlane so loops are scalar-branch clean and
// load offsets fold into immediates.
// ---------------------------------------------------------------------------
__global__ __launch_bounds__(128) void lap_gemm(const float* __restrict__ L,
                                                const float* __restrict__ xT,
                                                float* __restrict__ parts,
                                                int nchunks, int steps_per_chunk) {
    const int lane  = threadIdx.x & 31;
    const int unitV = blockIdx.x * (blockDim.x >> 5) + (threadIdx.x >> 5);
    const int unit  = __builtin_amdgcn_readfirstlane(unitV);   // wave-uniform -> SGPR
    if (unit >= NT * nchunks) return;

    const int tile  = unit % NT;             // tile-fastest: concurrent waves
    const int chunk = unit / NT;             // share the same xT k-window (L2)
    const int s0 = chunk * steps_per_chunk;
    int s1 = s0 + steps_per_chunk;
    if (s1 > KSTEPS) s1 = KSTEPS;
    const int sMain = (s1 < KFULL) ? s1 : KFULL;

    const int i = lane & 15;                 // 16-dim index (n for B, j for A)
    const int h = lane >> 4;                 // k half: lanes 16-31 hold k=2,3

    int n = tile * 16 + i;
    if (n > NV - 1) n = NV - 1;              // clamp; garbage cols masked later

    const float* pL = L  + (size_t)n * NV   + 4 * s0 + 2 * h;
    const float* pX = xT + (size_t)i * KPAD + 4 * s0 + 2 * h;

    v8f acc = {0.f, 0.f, 0.f, 0.f, 0.f, 0.f, 0.f, 0.f};

    const int nsteps = sMain - s0;           // scalar
    const int nbatch = nsteps >> 3;          // U = 8 steps per batch
    const int rem    = nsteps & 7;

    v2f aC[8], bC[8], aN[8], bN[8];
    if (nbatch > 0) {                        // preload batch 0
#pragma unroll
        for (int u = 0; u < 8; ++u) {
            aC[u] = *(const v2f*)(pX + 4 * u);
            bC[u] = *(const v2f*)(pL + 4 * u);
        }
        pX += 32; pL += 32;
    }
    for (int bt = 0; bt < nbatch; ++bt) {
        const bool more = (bt + 1 < nbatch);
        if (more) {                          // prefetch next batch (16 x b64)
#pragma unroll
            for (int u = 0; u < 8; ++u) {
                aN[u] = *(const v2f*)(pX + 4 * u);
                bN[u] = *(const v2f*)(pL + 4 * u);
            }
            pX += 32; pL += 32;
        }
#pragma unroll
        for (int u = 0; u < 8; ++u)          // consume current batch
            acc = WMMA_F32X4(aC[u], bC[u], acc);
        if (more) {
#pragma unroll
            for (int u = 0; u < 8; ++u) { aC[u] = aN[u]; bC[u] = bN[u]; }
        }
    }

    for (int t2 = 0; t2 < rem; ++t2) {       // leftover full steps (< 8)
        v2f a = *(const v2f*)(pX);
        v2f b = *(const v2f*)(pL);
        pX += 4; pL += 4;
        acc = WMMA_F32X4(a, b, acc);
    }

    if (s1 > KFULL) {                        // K tail: m = 10240..10243
        v2f a = *(const v2f*)(pX);           // xT is zero-padded to KPAD
        v2f b; b.x = 0.f; b.y = 0.f;
        if (h == 0) b = *(const v2f*)(pL);   // only k=0,1 in range
        acc = WMMA_F32X4(a, b, acc);
    }

    // Store raw partial D tile: element index = r*32 + lane.
    float* dst = parts + ((size_t)tile * nchunks + chunk) * 256 + lane;
#pragma unroll
    for (int r = 0; r < 8; ++r) dst[r * 32] = acc[r];
}

// ---------------------------------------------------------------------------
// Deterministic finish: sum k-chunk partials, square, mask invalid (j,n),
// reduce to out[0..3]. Single workgroup; fixed accumulation order.
// D layout: lane l, VGPR r  ->  row M = 8*(l/16) + r (= bd index),
//                               col N = l%16        (= n within tile).
// ---------------------------------------------------------------------------
__global__ __launch_bounds__(1024) void lap_reduce(const float* __restrict__ parts,
                                                   float* __restrict__ out,
                                                   int nchunks) {
    const int t = threadIdx.x;               // 0..1023
    const int e = t & 255;                   // element within a D tile
    const int q = t >> 8;                    // tile stride group 0..3
    const int l = e & 31;
    const int r = e >> 5;
    const int M = ((l >> 4) << 3) + r;
    const int N = l & 15;
    const int b = (M < 12) ? (M / 3) : -1;

    float acc[NBAT] = {0.f, 0.f, 0.f, 0.f};
    for (int T = q; T < NT; T += 4) {
        if (b >= 0 && (T * 16 + N) < NV) {
            const float* p = parts + (size_t)T * nchunks * 256 + e;
            float s = 0.f;
            for (int c = 0; c < nchunks; ++c) s += p[c * 256];
            acc[b] += s * s;
        }
    }

    __shared__ float red[1024];
    for (int bb = 0; bb < NBAT; ++bb) {
        red[t] = acc[bb];
        __syncthreads();
        for (int off = 512; off > 0; off >>= 1) {
            if (t < off) red[t] += red[t + off];
            __syncthreads();
        }
        if (t == 0) out[bb] = red[0];
        __syncthreads();
    }
}

// ---------------------------------------------------------------------------
// Fallback path (workspace too small): one wave per n-tile over full K,
// x read directly (masked), wave-shuffle reduction + atomicAdd.
// ---------------------------------------------------------------------------
__global__ void zero_out(float* __restrict__ out) {
    if (threadIdx.x < NBAT) out[threadIdx.x] = 0.0f;
}

__global__ __launch_bounds__(128) void lap_full(const float* __restrict__ L,
                                                const float* __restrict__ x,
                                                float* __restrict__ out) {
    const int lane  = threadIdx.x & 31;
    const int tileV = blockIdx.x * (blockDim.x >> 5) + (threadIdx.x >> 5);
    const int tile  = __builtin_amdgcn_readfirstlane(tileV);
    if (tile >= NT) return;

    const int i = lane & 15;
    const int h = lane >> 4;
    int n = tile * 16 + i;
    if (n > NV - 1) n = NV - 1;
    const float* rowL = L + (size_t)n * NV;

    const bool jval = (i < 12);
    const int jb = i / 3;
    const int jd = i - 3 * jb;
    const float* xr = x + (size_t)jb * NV * 3 + jd;

    v8f acc = {0.f, 0.f, 0.f, 0.f, 0.f, 0.f, 0.f, 0.f};
    for (int s = 0; s < KFULL; ++s) {
        const int m = 4 * s + 2 * h;
        v2f a;
        a.x = jval ? xr[(size_t)m * 3] : 0.f;
        a.y = jval ? xr[(size_t)(m + 1) * 3] : 0.f;
        v2f b = *(const v2f*)(rowL + m);
        acc = WMMA_F32X4(a, b, acc);
    }
    {   // tail step
        const int m = 4 * KFULL + 2 * h;
        v2f a;
        a.x = (jval && m     < NV) ? xr[(size_t)m * 3]       : 0.f;
        a.y = (jval && m + 1 < NV) ? xr[(size_t)(m + 1) * 3] : 0.f;
        v2f b; b.x = 0.f; b.y = 0.f;
        if (h == 0) b = *(const v2f*)(rowL + m);
        acc = WMMA_F32X4(a, b, acc);
    }

    float pb[NBAT] = {0.f, 0.f, 0.f, 0.f};
    if ((tile * 16 + i) < NV) {
#pragma unroll
        for (int r = 0; r < 8; ++r) {
            int M = 8 * h + r;
            if (M < 12) { float v = acc[r]; pb[M / 3] += v * v; }
        }
    }
#pragma unroll
    for (int bb = 0; bb < NBAT; ++bb) {
        float v = pb[bb];
        for (int off = 16; off > 0; off >>= 1) v += __shfl_xor(v, off, 32);
        if (lane == 0) atomicAdd(out + bb, v);
    }
}

// ---------------------------------------------------------------------------
extern "C" void kernel_launch(void* const* d_in, const int* in_sizes, int n_in,
                              void* d_out, int out_size, void* d_ws, size_t ws_size,
                              hipStream_t stream) {
    const float* x   = (const float*)d_in[0];   // (4, 10242, 3) fp32
    const float* Lap = (const float*)d_in[1];   // (10242, 10242) fp32
    float* out = (float*)d_out;                 // (4,) fp32

    const size_t xT_bytes = (size_t)16 * KPAD * sizeof(float);
    int C = 8;                                  // k-chunks -> 5128 waves
    while (C > 0 && ws_size < xT_bytes + (size_t)NT * C * 256 * sizeof(float)) --C;

    if (C > 0) {
        float* xT    = (float*)d_ws;
        float* parts = (float*)((char*)d_ws + xT_bytes);

        const int txN = 16 * KPAD;
        xT_build<<<(txN + 255) / 256, 256, 0, stream>>>(x, xT);

        const int steps = (KSTEPS + C - 1) / C;
        const int units = NT * C;
        lap_gemm<<<(units + 3) / 4, 128, 0, stream>>>(Lap, xT, parts, C, steps);
        lap_reduce<<<1, 1024, 0, stream>>>(parts, out, C);
    } else {
        zero_out<<<1, 32, 0, stream>>>(out);
        lap_full<<<(NT + 3) / 4, 128, 0, stream>>>(Lap, x, out);
    }
}